// ASBIGCN_75024488726907
// MI455X (gfx1250) — compile-verified
//
#include <hip/hip_runtime.h>
#include <hip/hip_bf16.h>

// ---------------------------------------------------------------------------
// HGAT-style attention (ASBIGCN reference) for MI455X / gfx1250, wave32.
// Bandwidth-bound on adj (340 MB); all matmuls via V_WMMA_F32_16X16X4_F32.
// ---------------------------------------------------------------------------

#define NTYPE 3
#define NNODE 3072
#define D_IN  512
#define D_OUT 64
#define GAMMA 0.1f
#define ONE_MINUS_GAMMA 0.9f

typedef float v2f __attribute__((ext_vector_type(2)));
typedef float v8f __attribute__((ext_vector_type(8)));

// ---------------------------------------------------------------------------
// Kernel 1: per-type projection  h[t] = x[t] @ W[t]   ([3072,512]@[512,64])
// 1 wave per 16x16 output tile; K-loop in steps of 4 (f32 WMMA).
// A-layout (16x4 f32): lane l -> row l&15; VGPR0 holds K = 2*(l>>4),
// VGPR1 holds K = 2*(l>>4)+1.  B (4x16): same K mapping, col = l&15.
// D (16x16): lane l -> col l&15, VGPR v -> row v + 8*(l>>4).
// ---------------------------------------------------------------------------
__global__ void proj_wmma_kernel(const float* __restrict__ x,
                                 const float* __restrict__ W,
                                 float* __restrict__ h) {
    int bid   = blockIdx.x;
    int ftile = bid % (D_OUT / 16); bid /= (D_OUT / 16);
    int ntile = bid % (NNODE / 16); bid /= (NNODE / 16);
    int t     = bid;

    int lane  = threadIdx.x & 31;
    int r     = lane & 15;     // A row / B col / D col
    int kh    = lane >> 4;     // K-half select

    const float* xrow = x + ((size_t)t * NNODE + ntile * 16 + r) * D_IN;
    const float* Wt   = W + (size_t)t * D_IN * D_OUT + ftile * 16;

    v8f acc = {};
    for (int k = 0; k < D_IN; k += 4) {
        int ka = k + 2 * kh;
        v2f a, b;
        a.x = xrow[ka];
        a.y = xrow[ka + 1];
        b.x = Wt[(size_t)ka * D_OUT + r];
        b.y = Wt[(size_t)(ka + 1) * D_OUT + r];
        acc = __builtin_amdgcn_wmma_f32_16x16x4_f32(
            false, a, false, b, (short)0, acc, false, false);
    }

    float* hout = h + ((size_t)t * NNODE + ntile * 16) * D_OUT + ftile * 16 + r;
#pragma unroll
    for (int v = 0; v < 8; ++v)
        hout[(size_t)(v + 8 * kh) * D_OUT] = acc[v];
}

// ---------------------------------------------------------------------------
// Kernel 2: attention logit vectors
//   s1[p,n]   = h[p,n,:] . a1[p]
//   s2[p,q,m] = h[q,m,:] . a2[p]
// ---------------------------------------------------------------------------
__global__ void attvec_kernel(const float* __restrict__ h,
                              const float* __restrict__ a1,
                              const float* __restrict__ a2,
                              float* __restrict__ s1,
                              float* __restrict__ s2) {
    int idx = blockIdx.x * blockDim.x + threadIdx.x;
    const int total1 = NTYPE * NNODE;
    if (idx < total1) {
        int p = idx / NNODE;
        const float* hp = h + (size_t)idx * D_OUT;
        const float* ap = a1 + p * D_OUT;
        float acc = 0.f;
#pragma unroll 8
        for (int f = 0; f < D_OUT; ++f) acc += hp[f] * ap[f];
        s1[idx] = acc;
    }
    int idx2 = idx - total1;
    if (idx2 >= 0 && idx2 < NTYPE * NTYPE * NNODE) {
        int m  = idx2 % NNODE;
        int pq = idx2 / NNODE;
        int q  = pq % NTYPE;
        int p  = pq / NTYPE;
        const float* hq = h + ((size_t)q * NNODE + m) * D_OUT;
        const float* ap = a2 + p * D_OUT;
        float acc = 0.f;
#pragma unroll 8
        for (int f = 0; f < D_OUT; ++f) acc += hq[f] * ap[f];
        s2[idx2] = acc;
    }
}

// ---------------------------------------------------------------------------
// Kernel 3: per-row softmax statistics. One wave per (p,q,n) row.
//   rowsum = sum_m adj ;  S = sum_{adj>0} exp(leakyrelu(s1+s2[m]))
//   c1 = GAMMA * rowsum / S   (0 if empty row)
// No max-shift needed: logit std ~5 keeps exp() well inside f32 range.
// ---------------------------------------------------------------------------
__global__ void rowstats_kernel(const float* __restrict__ adj,
                                const float* __restrict__ s1,
                                const float* __restrict__ s2,
                                float* __restrict__ c1) {
    int gwave = (int)((blockIdx.x * blockDim.x + threadIdx.x) >> 5);
    int lane  = threadIdx.x & 31;
    if (gwave >= NTYPE * NTYPE * NNODE) return;

    int n  = gwave % NNODE;
    int pq = gwave / NNODE;
    int p  = pq / NTYPE;

    const float* arow  = adj + (size_t)gwave * NNODE;
    const float* s2row = s2 + (size_t)pq * NNODE;
    float s1v = s1[p * NNODE + n];

    float rowsum = 0.f, S = 0.f;
    for (int m = lane; m < NNODE; m += 32) {
        float a = arow[m];
        if (a > 0.f) {
            rowsum += a;
            float z = s1v + s2row[m];
            float e = (z > 0.f) ? z : 0.2f * z;
            S += __expf(e);
        }
    }
#pragma unroll
    for (int off = 16; off > 0; off >>= 1) {
        rowsum += __shfl_down(rowsum, off, 32);
        S      += __shfl_down(S, off, 32);
    }
    if (lane == 0)
        c1[gwave] = (S > 0.f) ? (GAMMA * rowsum / S) : 0.f;
}

// ---------------------------------------------------------------------------
// Kernel 4: fused attention + aggregation
//   out[p,q,n,f] = sum_m adj*(c1*exp(lrelu(s1+s2[m])) + 0.9) * h[q,m,f] + bias
// Block = 128 threads (4 waves) on one 16-row n-tile; wave w owns cols
// [16w,16w+16). att A-tile built on the fly in WMMA f32 A-layout; all-zero
// tiles skipped wave-uniformly (EXEC stays all-ones for WMMA legality).
// ---------------------------------------------------------------------------
__global__ void aggregate_wmma_kernel(const float* __restrict__ adj,
                                      const float* __restrict__ h,
                                      const float* __restrict__ s1,
                                      const float* __restrict__ s2,
                                      const float* __restrict__ c1,
                                      const float* __restrict__ bias,
                                      float* __restrict__ out) {
    int bid   = blockIdx.x;
    int ntile = bid % (NNODE / 16);
    int pq    = bid / (NNODE / 16);
    int q     = pq % NTYPE;
    int p     = pq / NTYPE;

    int wave = threadIdx.x >> 5;
    int lane = threadIdx.x & 31;
    int r    = lane & 15;
    int kh   = lane >> 4;
    int f0   = wave * 16;
    int n0   = ntile * 16;

    const float* arow  = adj + ((size_t)pq * NNODE + n0 + r) * NNODE; // att row n0+r
    const float* s2row = s2 + (size_t)pq * NNODE;
    const float* hq    = h + (size_t)q * NNODE * D_OUT + f0 + r;      // B col f0+r
    float s1v = s1[p * NNODE + n0 + r];
    float c1v = c1[(size_t)pq * NNODE + n0 + r];

    v8f acc = {};
    for (int m0 = 0; m0 < NNODE; m0 += 4) {
        int ka = m0 + 2 * kh;
        float ad0 = arow[ka];
        float ad1 = arow[ka + 1];
        // wave-uniform skip of all-zero 16x4 att tiles (~27% at 2% density)
        if (__ballot(ad0 != 0.f || ad1 != 0.f) == 0ull) continue;

        float z0 = s1v + s2row[ka];
        float z1 = s1v + s2row[ka + 1];
        float e0 = (z0 > 0.f) ? z0 : 0.2f * z0;
        float e1 = (z1 > 0.f) ? z1 : 0.2f * z1;

        v2f a, b;
        a.x = (ad0 != 0.f) ? (c1v * __expf(e0) + ONE_MINUS_GAMMA) : 0.f;
        a.y = (ad1 != 0.f) ? (c1v * __expf(e1) + ONE_MINUS_GAMMA) : 0.f;
        b.x = hq[(size_t)ka * D_OUT];
        b.y = hq[(size_t)(ka + 1) * D_OUT];
        acc = __builtin_amdgcn_wmma_f32_16x16x4_f32(
            false, a, false, b, (short)0, acc, false, false);
    }

    float bv = bias[f0 + r];
    float* orow = out + ((size_t)pq * NNODE + n0) * D_OUT + f0 + r;
#pragma unroll
    for (int v = 0; v < 8; ++v)
        orow[(size_t)(v + 8 * kh) * D_OUT] = acc[v] + bv;
}

// ---------------------------------------------------------------------------
// Host launcher
// inputs: x[3,3072,512], adj[3,3,3072,3072], W[3,512,64], a1[3,64], a2[3,64],
//         bias[64]    -> out[3,3,3072,64]  (all f32)
// ---------------------------------------------------------------------------
extern "C" void kernel_launch(void* const* d_in, const int* in_sizes, int n_in,
                              void* d_out, int out_size, void* d_ws, size_t ws_size,
                              hipStream_t stream) {
    (void)in_sizes; (void)n_in; (void)out_size; (void)ws_size;
    const float* x    = (const float*)d_in[0];
    const float* adj  = (const float*)d_in[1];
    const float* W    = (const float*)d_in[2];
    const float* a1   = (const float*)d_in[3];
    const float* a2   = (const float*)d_in[4];
    const float* bias = (const float*)d_in[5];
    float* out = (float*)d_out;

    // workspace carve-up (all f32): h | s1 | s2 | c1   (~2.6 MB total)
    float* h  = (float*)d_ws;
    float* s1 = h  + (size_t)NTYPE * NNODE * D_OUT;
    float* s2 = s1 + (size_t)NTYPE * NNODE;
    float* c1 = s2 + (size_t)NTYPE * NTYPE * NNODE;

    // 1) projection: 3 * (3072/16) * (64/16) waves
    {
        dim3 grid(NTYPE * (NNODE / 16) * (D_OUT / 16));
        proj_wmma_kernel<<<grid, 32, 0, stream>>>(x, W, h);
    }
    // 2) s1 / s2 vectors
    {
        int total = NTYPE * NNODE + NTYPE * NTYPE * NNODE; // 36864
        dim3 grid((total + 255) / 256);
        attvec_kernel<<<grid, 256, 0, stream>>>(h, a1, a2, s1, s2);
    }
    // 3) row softmax stats: one wave per (p,q,n) row, 8 waves/block
    {
        int rows = NTYPE * NTYPE * NNODE; // 27648
        dim3 grid(rows / 8);
        rowstats_kernel<<<grid, 256, 0, stream>>>(adj, s1, s2, c1);
    }
    // 4) fused attention + aggregation: 9 * 192 blocks of 4 waves
    {
        dim3 grid(NTYPE * NTYPE * (NNODE / 16));
        aggregate_wmma_kernel<<<grid, 128, 0, stream>>>(adj, h, s1, s2, c1,
                                                        bias, out);
    }
}